// GammaLogProbCalculatorConstantDelta_89902255440405
// MI455X (gfx1250) — compile-verified
//
#include <hip/hip_runtime.h>
#include <cstdint>

// ---------------------------------------------------------------------------
// GammaLogProbCalculatorConstantDelta for MI455X (gfx1250)
//
// Memory-bound streaming kernel (~212 MB in, 1 float out, ~1 flop/byte).
// Strategy: Tensor Data Mover (TDM) DMA of contiguous 128-row cov_times tiles
// into LDS, double-buffered and tracked with TENSORcnt, compute overlapped;
// deterministic two-pass mean reduction via per-tile partials in d_ws.
// ---------------------------------------------------------------------------

#define N_TERMS        20
#define T_COV          50
#define ROWS_PER_TILE  128
#define TILE_ELEMS     (ROWS_PER_TILE * T_COV)   // 6400 floats, 25600 bytes
#define BLOCK_THREADS  128                       // 4 wave32s
#define MAX_GRID       1024

typedef __attribute__((ext_vector_type(4))) unsigned int u32x4;
typedef __attribute__((ext_vector_type(8))) int          i32x8;
typedef __attribute__((ext_vector_type(4))) int          i32x4;

// Build a 1-D TDM descriptor and issue TENSOR_LOAD_TO_LDS.
//   g0: [1:0] count=1 | [63:32] lds byte addr | [120:64] global addr | [127:126] type=2
//   g1: [17:16] data_size=2 (4B) | [79:48] tensor_dim0 (valid elems -> OOB zero-fill)
//       [111:80] tensor_dim1=1 | [127:112] tile_dim0 | [143:128] tile_dim1=1
//       [207:160] tensor_dim0_stride
__device__ __forceinline__ void tdm_load_tile(const float* gsrc,
                                              uint32_t lds_byte_addr,
                                              uint32_t valid_elems) {
  uint64_t ga = (uint64_t)(uintptr_t)gsrc;
  u32x4 g0;
  g0[0] = 1u;                                    // count=1, user mode, no gather
  g0[1] = lds_byte_addr;                         // LDS destination (bytes)
  g0[2] = (uint32_t)ga;                          // global_addr[31:0]
  g0[3] = (uint32_t)(ga >> 32) | (2u << 30);     // global_addr[56:32] | type=2

  i32x8 g1;
  g1[0] = (int)(2u << 16);                                     // data_size=4B, wg_mask=0
  g1[1] = (int)((valid_elems & 0xFFFFu) << 16);                // tensor_dim0[15:0]
  g1[2] = (int)(((valid_elems >> 16) & 0xFFFFu) | (1u << 16)); // tensor_dim0[31:16], tensor_dim1=1
  g1[3] = (int)((uint32_t)TILE_ELEMS << 16);                   // tensor_dim1[31:16]=0, tile_dim0
  g1[4] = 1;                                                   // tile_dim1=1, tile_dim2=0
  g1[5] = TILE_ELEMS;                                          // tensor_dim0_stride[31:0]
  g1[6] = 0;
  g1[7] = 0;

  i32x4 z4 = {0, 0, 0, 0};                       // groups 2/3 unused (<=2D tensor)
  i32x8 z8 = {0, 0, 0, 0, 0, 0, 0, 0};           // extra group (clang-23 6-arg form)
  __builtin_amdgcn_tensor_load_to_lds(g0, g1, z4, z4, z8, /*cpol=*/0);
}

// 20-term series for the lower regularized incomplete gamma P(a, x), x > 0:
//   P(a,x) = x^a e^-x / Gamma(a) * sum_{i=0..19} x^i / (a (a+1) ... (a+i))
__device__ __forceinline__ float lower_inc_gamma_p(float a, float x, float lgam) {
  float term = __fdividef(1.0f, a);
  float s = term;
#pragma unroll
  for (int j = 1; j < N_TERMS; ++j) {
    term *= __fdividef(x, a + (float)j);
    s += term;
  }
  return __expf(a * __logf(x) - x - lgam) * s;
}

__global__ void gamma_logprob_partial(const float* __restrict__ deltas,
                                      const float* __restrict__ event_times,
                                      const float* __restrict__ cov_times,
                                      const unsigned char* __restrict__ censor,
                                      const float* __restrict__ theta,
                                      float* __restrict__ partial,
                                      int nrows, int ntiles) {
  __shared__ float smem[2 * TILE_ELEMS + BLOCK_THREADS];
  float* buf0 = smem;
  float* buf1 = smem + TILE_ELEMS;
  float* red  = smem + 2 * TILE_ELEMS;

  const int  tid    = threadIdx.x;
  const bool wave0  = (tid >> 5) == 0;                  // uniform per wave32
  const long long totalElems = (long long)nrows * T_COV;

  const float alpha   = theta[0];
  const float beta    = theta[1];
  const float lgam    = lgammaf(alpha);
  const float logbeta = __logf(beta);

  const uint32_t lds0 = (uint32_t)(uintptr_t)buf0;      // flat LDS addr truncates to offset
  const uint32_t lds1 = (uint32_t)(uintptr_t)buf1;

  // Prologue: DMA first tile into buffer 0 (wave 0 only issues the TDM op).
  const int t0 = blockIdx.x;
  if (wave0) {
    long long base = (long long)t0 * TILE_ELEMS;
    long long rem  = totalElems - base;
    uint32_t valid = (uint32_t)(rem < (long long)TILE_ELEMS ? rem : (long long)TILE_ELEMS);
    tdm_load_tile(cov_times + base, lds0, valid);
  }

  int buf = 0;
  for (int t = t0; t < ntiles; t += gridDim.x) {
    const int nxt = t + gridDim.x;
    if (wave0) {
      if (nxt < ntiles) {
        // Prefetch next tile into the other buffer, then wait for tile t
        // (TDM ops from one wave complete in order -> tensorcnt<=1 means t done).
        long long base = (long long)nxt * TILE_ELEMS;
        long long rem  = totalElems - base;
        uint32_t valid = (uint32_t)(rem < (long long)TILE_ELEMS ? rem : (long long)TILE_ELEMS);
        tdm_load_tile(cov_times + base, buf ? lds0 : lds1, valid);
        __builtin_amdgcn_s_wait_tensorcnt(1);
      } else {
        __builtin_amdgcn_s_wait_tensorcnt(0);
      }
    }
    __syncthreads();   // publish tile t's LDS data to all 4 waves

    const float* tile = buf ? buf1 : buf0;
    const int r = t * ROWS_PER_TILE + tid;
    float lp = 0.0f;
    if (r < nrows) {
      // Row max over 50 floats; stride-50 over 64 banks is conflict-free for 32 lanes.
      float m = tile[tid * T_COV];
#pragma unroll
      for (int j = 1; j < T_COV; ++j) m = fmaxf(m, tile[tid * T_COV + j]);

      const float d   = deltas[r];
      const float set = event_times[r] + d;        // shifted event time
      const float sct = m + d;                     // shifted max covariate time

      const float logpdf = alpha * logbeta - lgam
                         + (alpha - 1.0f) * __logf(set) - beta * set;
      const float surv = 1.0f - lower_inc_gamma_p(alpha, set * beta, lgam);
      const float norm = 1.0f - lower_inc_gamma_p(alpha, sct * beta, lgam);
      lp = (censor[r] ? surv : logpdf) - norm;
    }

    // Deterministic per-tile tree reduction.
    red[tid] = lp;
    __syncthreads();
#pragma unroll
    for (int s = BLOCK_THREADS / 2; s > 0; s >>= 1) {
      if (tid < s) red[tid] += red[tid + s];
      __syncthreads();
    }
    if (tid == 0) partial[t] = red[0];
    __syncthreads();   // also fences LDS buffer reuse before next prefetch lands
    buf ^= 1;
  }
}

__global__ void reduce_mean(const float* __restrict__ partial, int ntiles,
                            float inv_n, float* __restrict__ out) {
  __shared__ float red[256];
  const int tid = threadIdx.x;
  float s = 0.0f;
  for (int i = tid; i < ntiles; i += 256) s += partial[i];  // fixed order: deterministic
  red[tid] = s;
  __syncthreads();
#pragma unroll
  for (int k = 128; k > 0; k >>= 1) {
    if (tid < k) red[tid] += red[tid + k];
    __syncthreads();
  }
  if (tid == 0) out[0] = red[0] * inv_n;
}

extern "C" void kernel_launch(void* const* d_in, const int* in_sizes, int n_in,
                              void* d_out, int out_size, void* d_ws, size_t ws_size,
                              hipStream_t stream) {
  const float*         deltas      = (const float*)d_in[0];         // (N,1) f32
  const float*         event_times = (const float*)d_in[1];         // (N,)  f32
  const float*         cov_times   = (const float*)d_in[2];         // (N,50) f32
  const unsigned char* censor      = (const unsigned char*)d_in[3]; // (N,) bool (1 byte)
  const float*         theta       = (const float*)d_in[4];         // (2,)  f32

  const int nrows  = in_sizes[1];
  const int ntiles = (nrows + ROWS_PER_TILE - 1) / ROWS_PER_TILE;
  float* partial = (float*)d_ws;   // ntiles floats of scratch

  const int grid = ntiles < MAX_GRID ? ntiles : MAX_GRID;
  gamma_logprob_partial<<<grid, BLOCK_THREADS, 0, stream>>>(
      deltas, event_times, cov_times, censor, theta, partial, nrows, ntiles);
  reduce_mean<<<1, 256, 0, stream>>>(partial, ntiles, 1.0f / (float)nrows,
                                     (float*)d_out);
}